// TtAttention_6004364280760
// MI455X (gfx1250) — compile-verified
//
#include <hip/hip_runtime.h>
#include <hip/hip_bf16.h>
#include <stdint.h>

typedef float v2f __attribute__((ext_vector_type(2)));
typedef float v8f __attribute__((ext_vector_type(8)));
typedef unsigned int u32x4 __attribute__((ext_vector_type(4)));
typedef int i32x4 __attribute__((ext_vector_type(4)));
typedef int i32x8 __attribute__((ext_vector_type(8)));

#define DIM      4096
#define NH       32
#define NKV      8
#define RG       4          // heads per kv group
#define HD       128
#define WIN      4096
#define BATCH    32
#define QKV_COLS 6144       // (32 + 2*8) * 128
#define KSPLIT   4

#define HAVE_TDM __has_builtin(__builtin_amdgcn_tensor_load_to_lds)

__device__ __forceinline__ v8f wmma4(v2f a, v2f b, v8f c) {
    // D(16x16,f32) = A(16x4,f32) x B(4x16,f32) + C
    return __builtin_amdgcn_wmma_f32_16x16x4_f32(
        false, a, false, b, (short)0, c, false, false);
}

#if HAVE_TDM
__device__ __forceinline__ unsigned lds_off(const void* p) {
    return (unsigned)(uintptr_t)(__attribute__((address_space(3))) const void*)p;
}

// 1-D TDM copy: nelem f32 elements from global to LDS byte offset lds_byte.
// D# group0: count=1 | lds_addr | 57-bit global addr | type=2 ("image")
// D# group1: data_size=4B, tensor_dim0=nelem, tensor_dim1=1,
//            tile_dim0=nelem, tile_dim1=1, dim0_stride=nelem
__device__ __forceinline__ void tdm_load_1d(unsigned lds_byte, const float* gsrc,
                                            unsigned nelem) {
    uint64_t ga = (uint64_t)(uintptr_t)gsrc;
    u32x4 g0;
    g0[0] = 1u;                                            // count=1 (valid user D#)
    g0[1] = lds_byte;                                      // lds_addr
    g0[2] = (unsigned)ga;                                  // global_addr[31:0]
    g0[3] = (unsigned)((ga >> 32) & 0x01FFFFFFu) | (2u << 30); // addr[56:32] | type=2
    i32x8 g1;
    g1[0] = (int)(2u << 16);                               // data_size = 4 bytes
    g1[1] = (int)((nelem & 0xFFFFu) << 16);                // tensor_dim0[15:0]
    g1[2] = (int)((nelem >> 16) | (1u << 16));             // dim0[31:16] | tensor_dim1=1
    g1[3] = (int)((nelem & 0xFFFFu) << 16);                // tile_dim0 = nelem
    g1[4] = 1;                                             // tile_dim1 = 1
    g1[5] = (int)nelem;                                    // tensor_dim0_stride lo32
    g1[6] = 0;
    g1[7] = 0;
    i32x4 z4 = {0, 0, 0, 0};
    i32x8 z8 = {0, 0, 0, 0, 0, 0, 0, 0};
    __builtin_amdgcn_tensor_load_to_lds(g0, g1, z4, z4, z8, 0);
}
#endif

// ---------------------------------------------------------------------------
// C(32 x N) = A(32 x K) @ W(K x N), K split KSPLIT ways into partials.
// One wave per (column-tile, k-slice): two 16x16 f32 WMMA accumulators.
// ---------------------------------------------------------------------------
__global__ __launch_bounds__(256)
void gemm32_part(const float* __restrict__ A, const float* __restrict__ W,
                 float* __restrict__ Cpart, int K, int N, int tiles) {
    const int lane = threadIdx.x & 31;
    const int wg   = blockIdx.x * (blockDim.x >> 5) + (threadIdx.x >> 5);
    const int tile = wg % tiles;
    const int ks   = wg / tiles;
    const int c0   = tile * 16;
    const int kchunk = K / KSPLIT;
    const int kbeg = ks * kchunk;
    const int kend = kbeg + kchunk;

    const bool hi  = lane >= 16;
    const int col  = c0 + (lane & 15);
    const int hsel = hi ? 2 : 0;            // which K-pair this half-wave owns
    const int rowa = lane & 15;

    v8f acc0 = {};
    v8f acc1 = {};

#pragma unroll 4
    for (int k = kbeg; k < kend; k += 4) {
        float4 fa0 = *(const float4*)(A + rowa * K + k);
        float4 fa1 = *(const float4*)(A + (rowa + 16) * K + k);
        v2f a0, a1, bb;
        a0.x = hi ? fa0.z : fa0.x;  a0.y = hi ? fa0.w : fa0.y;
        a1.x = hi ? fa1.z : fa1.x;  a1.y = hi ? fa1.w : fa1.y;
        bb.x = W[(k + hsel) * N + col];
        bb.y = W[(k + hsel + 1) * N + col];
        acc0 = wmma4(a0, bb, acc0);
        acc1 = wmma4(a1, bb, acc1);
    }

    float* Cp = Cpart + (size_t)ks * 32 * N;
    const int rbase = hi ? 8 : 0;
#pragma unroll
    for (int j = 0; j < 8; ++j) {
        Cp[(rbase + j) * N + col]      = acc0[j];
        Cp[(16 + rbase + j) * N + col] = acc1[j];
    }
}

// out[i] = sum over KSPLIT partial planes
__global__ void reduce4(const float* __restrict__ part, float* __restrict__ out, int n) {
    int i = blockIdx.x * blockDim.x + threadIdx.x;
    if (i < n)
        out[i] = part[i] + part[i + n] + part[i + 2 * n] + part[i + 3 * n];
}

// In-place RoPE on q (heads 0..31) and k (heads 32..39) columns of qkv ws.
__global__ void rope_kernel(float* __restrict__ qkv,
                            const float* __restrict__ cosv,
                            const float* __restrict__ sinv) {
    int idx = blockIdx.x * blockDim.x + threadIdx.x;   // BATCH * 40 * 64
    if (idx >= BATCH * 40 * 64) return;
    int i = idx & 63;
    int h = (idx >> 6) % 40;
    int b = idx / (40 * 64);
    float* row = qkv + b * QKV_COLS + h * HD;
    float t1 = row[i];
    float t2 = row[i + 64];
    row[i]      = t1 * cosv[i]      - t2 * sinv[i];
    row[i + 64] = t2 * cosv[i + 64] + t1 * sinv[i + 64];
}

// ---------------------------------------------------------------------------
// Flash-decode attention. One block per (batch, kv-group); 8 waves split the
// key window. K tiles are streamed into LDS by the Tensor Data Mover (double
// buffered, s_wait_tensorcnt), scores via V_WMMA_F32_16X16X4_F32, online
// softmax, A*V via coalesced b128 V loads (+ global_prefetch of next tile)
// with shfl p-broadcast. Cache slot `cur` is fixed up from the fresh k/v.
// ---------------------------------------------------------------------------
__global__ __launch_bounds__(256)
void attn_kernel(const float* __restrict__ qkvws,
                 const float* __restrict__ cache_k,
                 const float* __restrict__ cache_v,
                 const float* __restrict__ mask,
                 const int*   __restrict__ sp_ptr,
                 float* __restrict__ attn_out) {
    const int bg = blockIdx.x;
    const int b  = bg >> 3;
    const int g  = bg & 7;
    const int lane = threadIdx.x & 31;
    const int w    = threadIdx.x >> 5;
    const bool hi  = lane >= 16;

    const int sp  = sp_ptr[0];
    const int cur = sp % WIN;
    int plen = ((sp + 1 + 31) >> 5) << 5;
    if (plen > WIN) plen = WIN;

#if HAVE_TDM
    __shared__ __align__(16) float kbuf[8][2][16 * HD];   // 128 KB double buffers
#endif
    __shared__ __align__(16) float lds_acc[8][4][HD];
    __shared__ float lds_m[8][4];
    __shared__ float lds_l[8][4];

    const float* Kb   = cache_k + (size_t)(b * NKV + g) * WIN * HD;
    const float* Vb   = cache_v + (size_t)(b * NKV + g) * WIN * HD;
    const float* knew = qkvws + b * QKV_COLS + NH * HD + g * HD;
    const float* vnew = qkvws + b * QKV_COLS + (NH + NKV) * HD + g * HD;
    const float* mrow = mask + b * WIN;

    const int h = lane & 15;               // head slot this lane tracks (valid if < RG)
    const float scale = 0.088388347648318447f;  // 1/sqrt(128)

    // Preload Q as WMMA B-operand chunks: qb[i] covers d = 4i .. 4i+3
    v2f qb[32];
    {
        const float* qrow = qkvws + b * QKV_COLS + (g * RG + h) * HD + (hi ? 2 : 0);
        const bool valid = h < RG;
#pragma unroll
        for (int i = 0; i < 32; ++i) {
            qb[i].x = valid ? qrow[4 * i]     : 0.0f;
            qb[i].y = valid ? qrow[4 * i + 1] : 0.0f;
        }
    }

    float m_run = -3.0e38f;
    float l_run = 0.0f;
    float4 acc[RG];
#pragma unroll
    for (int hh = 0; hh < RG; ++hh) { acc[hh].x = acc[hh].y = acc[hh].z = acc[hh].w = 0.0f; }

    const int tbeg = w * (WIN / 8);
    int tend = tbeg + (WIN / 8);
    if (tend > plen) tend = plen;

#if HAVE_TDM
    // escape the LDS buffer so memory clobbers order ds reads vs TDM writes
    { void* esc = (void*)&kbuf[0][0][0]; asm volatile("" :: "v"(esc) : "memory"); }
    const unsigned kbase = lds_off(&kbuf[w][0][0]);
    if (tbeg < tend)
        tdm_load_1d(kbase, Kb + (size_t)tbeg * HD, 16 * HD);
#endif
    int nb = 0;

    for (int t0 = tbeg; t0 < tend; t0 += 16) {
#if HAVE_TDM
        if (t0 + 16 < tend) {   // issue next tile, wait for current
            tdm_load_1d(kbase + (unsigned)((nb ^ 1) * 16 * HD * 4),
                        Kb + (size_t)(t0 + 16) * HD, 16 * HD);
            __builtin_amdgcn_s_wait_tensorcnt(1);
        } else {
            __builtin_amdgcn_s_wait_tensorcnt(0);
        }
        asm volatile("" ::: "memory");
        const float* krow = &kbuf[w][nb][(lane & 15) * HD];
#else
        const float* krow = Kb + (size_t)(t0 + h) * HD;
#endif
        // prefetch next V tile (8 KB) into cache
        if (t0 + 16 < tend) {
            const float* vn = Vb + (size_t)(t0 + 16) * HD;
            __builtin_prefetch(vn + lane * 32, 0, 3);
            __builtin_prefetch(vn + 1024 + lane * 32, 0, 3);
        }

        // ---- scores: D[key m][head n] over 16 keys x 4 heads --------------
        v8f s = {};
#pragma unroll
        for (int i = 0; i < 32; ++i) {
            float4 kf = *(const float4*)(krow + 4 * i);
            v2f a;
            a.x = hi ? kf.z : kf.x;
            a.y = hi ? kf.w : kf.y;
            s = wmma4(a, qb[i], s);
        }
        float sv[8];
#pragma unroll
        for (int j = 0; j < 8; ++j) sv[j] = s[j] * scale;

        // replace stale score at the freshly written cache slot
        if (cur >= t0 && cur < t0 + 16) {
            int j = cur - t0;
            bool mine = (j < 8) ? !hi : hi;
            if (mine) {
                const float* qp = qkvws + b * QKV_COLS + (g * RG + h) * HD;
                float d0 = 0.0f;
                for (int d = 0; d < HD; ++d) d0 += qp[d] * knew[d];
                sv[j & 7] = d0 * scale;
            }
        }

        // additive mask
        {
            int mb = t0 + (hi ? 8 : 0);
            float4 m0 = *(const float4*)(mrow + mb);
            float4 m1 = *(const float4*)(mrow + mb + 4);
            sv[0] += m0.x; sv[1] += m0.y; sv[2] += m0.z; sv[3] += m0.w;
            sv[4] += m1.x; sv[5] += m1.y; sv[6] += m1.z; sv[7] += m1.w;
        }

        // ---- online softmax update ---------------------------------------
        float mt = sv[0];
#pragma unroll
        for (int j = 1; j < 8; ++j) mt = fmaxf(mt, sv[j]);
        mt = fmaxf(mt, __shfl_xor(mt, 16, 32));
        float mnew = fmaxf(m_run, mt);
        float corr = __expf(m_run - mnew);
        m_run = mnew;

        float pv[8];
        float psum = 0.0f;
#pragma unroll
        for (int j = 0; j < 8; ++j) { pv[j] = __expf(sv[j] - mnew); psum += pv[j]; }
        l_run = l_run * corr + psum;

        float corrs[RG];
#pragma unroll
        for (int hh = 0; hh < RG; ++hh) corrs[hh] = __shfl(corr, hh, 32);
#pragma unroll
        for (int hh = 0; hh < RG; ++hh) {
            acc[hh].x *= corrs[hh]; acc[hh].y *= corrs[hh];
            acc[hh].z *= corrs[hh]; acc[hh].w *= corrs[hh];
        }

        // ---- A*V: lane owns d = lane*4 .. lane*4+3 for all 4 heads -------
#pragma unroll
        for (int m = 0; m < 16; ++m) {
            int key = t0 + m;
            const float* vrow = (key == cur) ? vnew : (Vb + (size_t)key * HD);
            float4 vv = *(const float4*)(vrow + lane * 4);
            int srcbase = (m < 8) ? 0 : 16;
#pragma unroll
            for (int hh = 0; hh < RG; ++hh) {
                float p = __shfl(pv[m & 7], srcbase + hh, 32);
                acc[hh].x += p * vv.x; acc[hh].y += p * vv.y;
                acc[hh].z += p * vv.z; acc[hh].w += p * vv.w;
            }
        }
        nb ^= 1;
    }

    // ---- per-wave partials to LDS ----------------------------------------
    float l_tot = l_run + __shfl_xor(l_run, 16, 32);
    if (lane < RG) { lds_m[w][lane] = m_run; lds_l[w][lane] = l_tot; }
#pragma unroll
    for (int hh = 0; hh < RG; ++hh)
        *(float4*)&lds_acc[w][hh][lane * 4] = acc[hh];
    __syncthreads();

    // ---- combine 8 waves --------------------------------------------------
    for (int o = threadIdx.x; o < RG * HD; o += 256) {
        int hh = o >> 7;
        int d  = o & (HD - 1);
        float M = lds_m[0][hh];
#pragma unroll
        for (int w2 = 1; w2 < 8; ++w2) M = fmaxf(M, lds_m[w2][hh]);
        float L = 0.0f, val = 0.0f;
#pragma unroll
        for (int w2 = 0; w2 < 8; ++w2) {
            float e = __expf(lds_m[w2][hh] - M);
            L   += lds_l[w2][hh] * e;
            val += lds_acc[w2][hh][d] * e;
        }
        attn_out[b * (NH * HD) + (g * RG + hh) * HD + d] = val / L;
    }
}

// ---------------------------------------------------------------------------
extern "C" void kernel_launch(void* const* d_in, const int* in_sizes, int n_in,
                              void* d_out, int out_size, void* d_ws, size_t ws_size,
                              hipStream_t stream) {
    const float* x       = (const float*)d_in[0];
    const float* wqkv    = (const float*)d_in[1];
    const float* wo      = (const float*)d_in[2];
    const float* cosv    = (const float*)d_in[3];
    const float* sinv    = (const float*)d_in[4];
    const float* cache_k = (const float*)d_in[5];
    const float* cache_v = (const float*)d_in[6];
    const float* mask    = (const float*)d_in[7];
    const int*   sp      = (const int*)d_in[8];

    float* qkv      = (float*)d_ws;                          // 32*6144
    float* qkv_part = qkv + BATCH * QKV_COLS;                // 4 * 32*6144
    float* attn_out = qkv_part + KSPLIT * BATCH * QKV_COLS;  // 32*4096
    float* out_part = attn_out + BATCH * DIM;                // 4 * 32*4096

    // 1) QKV projection: 384 col-tiles * 4 k-slices = 1536 waves
    gemm32_part<<<192, 256, 0, stream>>>(x, wqkv, qkv_part, DIM, QKV_COLS, QKV_COLS / 16);
    reduce4<<<(BATCH * QKV_COLS) / 256, 256, 0, stream>>>(qkv_part, qkv, BATCH * QKV_COLS);

    // 2) RoPE on q + k (in place)
    rope_kernel<<<(BATCH * 40 * 64 + 255) / 256, 256, 0, stream>>>(qkv, cosv, sinv);

    // 3) Flash-decode attention: one block per (batch, kv-group)
    attn_kernel<<<BATCH * NKV, 256, 0, stream>>>(qkv, cache_k, cache_v, mask, sp, attn_out);

    // 4) Output projection: 256 col-tiles * 4 k-slices = 1024 waves
    gemm32_part<<<128, 256, 0, stream>>>(attn_out, wo, out_part, DIM, DIM, DIM / 16);
    reduce4<<<(BATCH * DIM) / 256, 256, 0, stream>>>(out_part, (float*)d_out, BATCH * DIM);
}